// PLIF_73538430042799
// MI455X (gfx1250) — compile-verified
//
#include <hip/hip_runtime.h>

// PLIF spiking forward: x:[B,T,C,H,W] f32, tau:[C] f32 -> spikes:[B,T,C,H,W] f32
// Memory-bound streaming kernel (AI ~0.5 FLOP/B; HBM floor ~11.5us @ 23.3 TB/s).
// One thread = one float4 of neurons (4 consecutive w). T-recurrence in registers;
// all 8 time-step loads issued up-front, nontemporal b128 loads/stores.

typedef float __attribute__((ext_vector_type(4))) f32x4;

#define B_  32
#define T_  8
#define C_  128
#define H_  32
#define W_  32
#define HW_          (H_ * W_)        // 1024
#define CHW_         (C_ * HW_)       // 131072 elements (512 KiB) per (b,t) image
#define VEC_PER_IMG  (CHW_ / 4)       // 32768 float4 vectors per (b,t)
#define THRESH_      1.0f

__global__ __launch_bounds__(256)
void plif_fwd_kernel(const float* __restrict__ x,
                     const float* __restrict__ tau,
                     float* __restrict__ out) {
    const int tid = blockIdx.x * blockDim.x + threadIdx.x;   // 0 .. B*VEC_PER_IMG-1
    const int b   = tid >> 15;                               // tid / 32768
    const int rv  = tid & (VEC_PER_IMG - 1);                 // vec index inside image
    const int c   = rv >> 8;                                 // rv / (HW/4); 256 threads (one block) share c
    const size_t r = (size_t)rv << 2;                        // element offset in image

    // sigmoid(tau[c]) once per thread (one v_exp_f32; broadcast via cache)
    const float tc    = tau[c];
    const float tau_s = 1.0f / (1.0f + __expf(-tc));

    const size_t base = (size_t)b * (size_t)T_ * (size_t)CHW_ + r;

    // Issue all 8 independent time-step loads before the recurrence (NT, b128).
    f32x4 xv[T_];
#pragma unroll
    for (int t = 0; t < T_; ++t) {
        xv[t] = __builtin_nontemporal_load(
            (const f32x4*)(x + base + (size_t)t * (size_t)CHW_));
    }

    f32x4 mem = {0.0f, 0.0f, 0.0f, 0.0f};
#pragma unroll
    for (int t = 0; t < T_; ++t) {
        f32x4 spk;
#pragma unroll
        for (int i = 0; i < 4; ++i) {
            float m = mem[i] * tau_s + xv[t][i];   // leak + integrate (v_fma_f32)
            bool  fired = (m > THRESH_);           // zif(mem - 1) == (mem > 1)
            spk[i] = fired ? 1.0f : 0.0f;
            mem[i] = fired ? 0.0f : m;             // hard reset
        }
        __builtin_nontemporal_store(spk, (f32x4*)(out + base + (size_t)t * (size_t)CHW_));
    }
}

extern "C" void kernel_launch(void* const* d_in, const int* in_sizes, int n_in,
                              void* d_out, int out_size, void* d_ws, size_t ws_size,
                              hipStream_t stream) {
    const float* x   = (const float*)d_in[0];   // [B,T,C,H,W]
    const float* tau = (const float*)d_in[1];   // [C]
    float* out       = (float*)d_out;           // [B,T,C,H,W]

    const int nthreads = B_ * VEC_PER_IMG;      // 1,048,576 threads
    const int block    = 256;                   // 8 wave32 per block
    const int grid     = nthreads / block;      // 4096 blocks

    plif_fwd_kernel<<<grid, block, 0, stream>>>(x, tau, out);
}